// interpolator_iwd_89060441849912
// MI455X (gfx1250) — compile-verified
//
#include <hip/hip_runtime.h>
#include <cstdint>
#include <cstddef>

// ---------------- problem constants (match reference) ----------------
#define B_     4
#define SDIM   4096
#define TDIM   4096
#define NHH    16
#define SPLIT  4                    // split S for occupancy: 256 WGs total
#define SCHUNK (SDIM / SPLIT)       // 1024
#define COLS   256                  // ti columns per workgroup (== blockDim)
#define TILE_S 8                    // s rows staged per LDS tile
#define NTILES (SCHUNK / TILE_S)    // 128

// ---------------- CDNA5 async global->LDS path (guarded) ----------------
#if defined(__has_builtin)
#if __has_builtin(__builtin_amdgcn_global_load_async_to_lds_b128)
#define HAVE_ASYNC 1
#endif
#endif
#ifndef HAVE_ASYNC
#define HAVE_ASYNC 0
#endif

#if !HAVE_ASYNC
#define WAIT_ASYNC(n) do { } while (0)
#elif defined(__has_builtin) && __has_builtin(__builtin_amdgcn_s_wait_asynccnt)
#define WAIT_ASYNC(n) __builtin_amdgcn_s_wait_asynccnt(n)
#else
#define WAIT_ASYNC(n) asm volatile("s_wait_asynccnt " #n ::: "memory")
#endif

// builtin parameter types: pointer-to-int4 in AS(1)/AS(3)
typedef int v4i_ __attribute__((__vector_size__(4 * sizeof(int))));
typedef __attribute__((address_space(1))) v4i_ as1_v4i;
typedef __attribute__((address_space(3))) v4i_ as3_v4i;

// WMMA vector types
typedef __attribute__((ext_vector_type(2))) float v2f;
typedef __attribute__((ext_vector_type(8))) float v8f;

// =====================================================================
// Kernel 1: per-column top-16 smallest d^2 over an S-chunk.
// grid = (TDIM/COLS, SPLIT, B_), block = COLS (256 threads = 8 waves).
// coords_rel streamed via double-buffered async global->LDS tiles.
// =====================================================================
__global__ __launch_bounds__(COLS) void k_topk_partial(
    const float* __restrict__ coords,   // [2][B][S][T]
    float* __restrict__ pvals,          // [B][SPLIT][T][NH]  (d^2, ascending)
    int*   __restrict__ pidx)           // [B][SPLIT][T][NH]  (s index)
{
    __shared__ __align__(16) float tile[2][2][TILE_S][COLS]; // [buf][chan][s][ti]

    const int tx    = threadIdx.x;
    const int tbase = blockIdx.x * COLS;
    const int sp    = blockIdx.y;
    const int bi    = blockIdx.z;
    const int s0    = sp * SCHUNK;
    const int ti    = tbase + tx;

    const size_t chanStride = (size_t)B_ * SDIM * TDIM;
    const float* base0 = coords + (size_t)bi * SDIM * TDIM + tbase; // chan 0
    const float* base1 = base0 + chanStride;                        // chan 1

    // Copy one 2ch x TILE_S x COLS tile (16 KB) into LDS buffer `buf`.
    // 1024 x 16B units; 256 threads x 4 rounds; per wave: 4 async instrs.
    auto issue_tile = [&](int tk, int buf) {
        const int srow_base = s0 + tk * TILE_S;
        #pragma unroll
        for (int r = 0; r < 4; ++r) {
            int u    = r * COLS + tx;     // 0..1023
            int row  = u >> 6;            // 0..15  (chan,srow)
            int col  = (u & 63) << 2;     // float col, step 4 (16B)
            int c    = row >> 3;
            int sr   = row & 7;
            const float* g = (c == 0 ? base0 : base1) +
                             (size_t)(srow_base + sr) * TDIM + col;
            float* l = &tile[buf][c][sr][col];
#if HAVE_ASYNC
            __builtin_amdgcn_global_load_async_to_lds_b128(
                (as1_v4i*)(uintptr_t)g,
                (as3_v4i*)(uintptr_t)(uint32_t)(uintptr_t)l,
                0, 0);
#else
            *(float4*)l = *(const float4*)g;
#endif
        }
    };

    // running top-16 (ascending) of d^2, fully register-resident
    float bv[NHH];
    int   bx[NHH];
    #pragma unroll
    for (int h = 0; h < NHH; ++h) { bv[h] = 3.4e38f; bx[h] = 0; }

    issue_tile(0, 0);

    for (int tk = 0; tk < NTILES; ++tk) {
        const int cur = tk & 1;
        if (tk + 1 < NTILES) {
            issue_tile(tk + 1, cur ^ 1);
            WAIT_ASYNC(4);          // current tile's 4 ops done; next 4 in flight
        } else {
            WAIT_ASYNC(0);
        }
        __syncthreads();            // everyone's slice of `cur` is visible

        const int sbase = s0 + tk * TILE_S;
        #pragma unroll
        for (int sl = 0; sl < TILE_S; ++sl) {
            float c0 = tile[cur][0][sl][tx];
            float c1 = tile[cur][1][sl][tx];
            float d2 = fmaf(c0, c0, c1 * c1);
            if (d2 < bv[NHH - 1]) {              // rare: ~105/4096 per column
                bv[NHH - 1] = d2;
                bx[NHH - 1] = sbase + sl;
                #pragma unroll
                for (int j = NHH - 1; j > 0; --j) {
                    if (bv[j] < bv[j - 1]) {     // strict < keeps ties stable
                        float tv = bv[j - 1]; bv[j - 1] = bv[j]; bv[j] = tv;
                        int   tx2 = bx[j - 1]; bx[j - 1] = bx[j]; bx[j] = tx2;
                    }
                }
            }
        }
        __syncthreads();            // before buffer `cur` is overwritten
    }

    const size_t pbase = (((size_t)bi * SPLIT + sp) * TDIM + ti) * NHH;
    #pragma unroll
    for (int h = 0; h < NHH; ++h) {
        pvals[pbase + h] = bv[h];
        pidx [pbase + h] = bx[h];
    }
}

// =====================================================================
// Kernel 2: merge SPLIT sorted 16-lists per column -> final top-16,
// compute w = 1/(sqrt(d2)+1e-15)^2, emit w[b][nh][t], idx[b][nh][t].
// =====================================================================
__global__ __launch_bounds__(256) void k_merge(
    const float* __restrict__ pvals, const int* __restrict__ pidx,
    float* __restrict__ wbuf, int* __restrict__ ibuf)
{
    const int g  = blockIdx.x * 256 + threadIdx.x;   // 0..B*T-1
    const int bi = g >> 12;
    const int ti = g & (TDIM - 1);
    const size_t cstride = (size_t)TDIM * NHH;
    const size_t base    = ((size_t)bi * SPLIT * TDIM + ti) * NHH;

    float hv[SPLIT]; int hx[SPLIT]; int hp[SPLIT];
    #pragma unroll
    for (int c = 0; c < SPLIT; ++c) {
        hp[c] = 0;
        hv[c] = pvals[base + c * cstride];
        hx[c] = pidx [base + c * cstride];
    }

    for (int k = 0; k < NHH; ++k) {
        int cb = 0; float bvv = hv[0];
        #pragma unroll
        for (int c = 1; c < SPLIT; ++c)
            if (hv[c] < bvv) { bvv = hv[c]; cb = c; }   // ties -> lower chunk (lower s)
        int xb = hx[0];
        #pragma unroll
        for (int c = 1; c < SPLIT; ++c) if (cb == c) xb = hx[c];

        float d = sqrtf(bvv);
        float t = d + 1e-15f;
        float w = 1.0f / (t * t);
        size_t o = ((size_t)bi * NHH + k) * TDIM + ti;
        wbuf[o] = w;
        ibuf[o] = xb;

        #pragma unroll
        for (int c = 0; c < SPLIT; ++c) {
            if (cb == c) {
                hp[c]++;
                bool ok = hp[c] < NHH;
                hv[c] = ok ? pvals[base + c * cstride + hp[c]] : 3.4e38f;
                hx[c] = ok ? pidx [base + c * cstride + hp[c]] : 0;
            }
        }
    }
}

// =====================================================================
// Kernel 3: denom[b][nh] = sum_t w[b][nh][t], as a WMMA reduction:
// D(16x16) += A(16x4 rows of w) x B(4x16 all-ones). Every column of D
// holds the row sum, so the B/D lane layouts are irrelevant.
// grid = B_ (one 16-row block per batch), block = 32 (one wave, EXEC all 1s).
// =====================================================================
__global__ __launch_bounds__(32) void k_denom(
    const float* __restrict__ wbuf, float* __restrict__ denom)
{
#if defined(__has_builtin) && __has_builtin(__builtin_amdgcn_wmma_f32_16x16x4_f32)
    const int rb    = blockIdx.x;       // row block == batch (NHH == 16)
    const int lane  = threadIdx.x;      // 0..31
    const int m     = lane & 15;        // A-matrix row for this lane
    const int khalf = lane >> 4;        // lanes 16..31 hold K=2,3
    const float* rowp = wbuf + ((size_t)rb * NHH + m) * TDIM + khalf * 2;

    v8f acc = {};
    v2f bones = {1.0f, 1.0f};
    for (int kc = 0; kc < TDIM / 4; ++kc) {
        v2f a; a.x = rowp[0]; a.y = rowp[1];
        rowp += 4;
        acc = __builtin_amdgcn_wmma_f32_16x16x4_f32(
            false, a, false, bones, (short)0, acc, false, false);
    }
    // D layout: VGPR j = row j (lanes 0-15) / row j+8 (lanes 16-31)
    #pragma unroll
    for (int j = 0; j < 8; ++j) {
        if (lane == j)      denom[rb * NHH + j]     = acc[j];
        if (lane == j + 16) denom[rb * NHH + 8 + j] = acc[j];
    }
#else
    const int rb   = blockIdx.x;
    const int lane = threadIdx.x;
    const int row  = rb * NHH + (lane & 15);
    const int half = lane >> 4;
    const float* rp = wbuf + (size_t)row * TDIM + half * (TDIM / 2);
    float s = 0.0f;
    for (int k = 0; k < TDIM / 2; ++k) s += rp[k];
    __shared__ float red[32];
    red[lane] = s;
    __syncthreads();
    if (lane < 16) denom[rb * NHH + lane] = red[lane] + red[lane + 16];
#endif
}

// =====================================================================
// Kernel 4: faithful .view() reinterpretation + gather + NH-sum.
// w2[bi,ti,h] = w_flat[bi*65536 + ti*16 + h] / denom[ti%4][h]
// out[bi,ti]  = sum_h x[bi, idx_flat[bi*65536 + ti*16 + h]] * w2
// =====================================================================
__global__ __launch_bounds__(256) void k_out(
    const float* __restrict__ x,        // [B][S][1]
    const float* __restrict__ wbuf,     // flat [B*NH*T]
    const int*   __restrict__ ibuf,     // flat [B*NH*T]
    const float* __restrict__ denom,    // [B][NH]
    float* __restrict__ out)            // [B][T][1]
{
    const int g  = blockIdx.x * 256 + threadIdx.x;
    const int bi = g >> 12;
    const int ti = g & (TDIM - 1);
    const size_t f = (size_t)bi * (NHH * TDIM) + (size_t)ti * NHH; // contiguous 16
    const float* wp = wbuf + f;
    const int*   ip = ibuf + f;
    const float* dq = denom + (ti & 3) * NHH;   // q = ti % 4 (reinterp bug, faithful)
    const float* xb = x + (size_t)bi * SDIM;

    float acc = 0.0f;
    #pragma unroll
    for (int h = 0; h < NHH; ++h) {
        float w = wp[h] / dq[h];
        acc = fmaf(xb[ip[h]], w, acc);
    }
    out[g] = acc;
}

// =====================================================================
extern "C" void kernel_launch(void* const* d_in, const int* in_sizes, int n_in,
                              void* d_out, int out_size, void* d_ws, size_t ws_size,
                              hipStream_t stream) {
    (void)in_sizes; (void)n_in; (void)out_size; (void)ws_size;
    const float* x      = (const float*)d_in[0];   // [4,4096,1]
    const float* coords = (const float*)d_in[1];   // [2,4,4096,4096]
    float* out = (float*)d_out;                    // [4,4096,1]

    char* ws = (char*)d_ws;                        // ~10 MB used
    float* pvals = (float*)(ws);                                  // 4 MB
    int*   pidx  = (int*)  (ws + (size_t) 4 * 1024 * 1024);       // 4 MB
    float* wbuf  = (float*)(ws + (size_t) 8 * 1024 * 1024);       // 1 MB
    int*   ibuf  = (int*)  (ws + (size_t) 9 * 1024 * 1024);       // 1 MB
    float* denom = (float*)(ws + (size_t)10 * 1024 * 1024);       // 256 B

    dim3 g1(TDIM / COLS, SPLIT, B_);
    hipLaunchKernelGGL(k_topk_partial, g1, dim3(COLS), 0, stream, coords, pvals, pidx);
    hipLaunchKernelGGL(k_merge, dim3((B_ * TDIM) / 256), dim3(256), 0, stream,
                       pvals, pidx, wbuf, ibuf);
    hipLaunchKernelGGL(k_denom, dim3(B_), dim3(32), 0, stream, wbuf, denom);
    hipLaunchKernelGGL(k_out, dim3((B_ * TDIM) / 256), dim3(256), 0, stream,
                       x, wbuf, ibuf, denom, out);
}